// Build_GPyr_49795850829871
// MI455X (gfx1250) — compile-verified
//
#include <hip/hip_runtime.h>

// ---------------------------------------------------------------------------
// Gaussian pyramid build, MI455X (gfx1250).
// Memory-bound (AI ~0.06 FLOP/B): total traffic ~530 MB -> ~23 us floor at
// 23.3 TB/s. Strategy:
//  - reduce-1 kernel FUSES the level-0 copy (writes it from the LDS tile),
//    removing a separate 201 MB read pass.
//  - haloed 67x19 tile staged into LDS with CDNA5 async global->LDS loads
//    (ASYNCcnt / s_wait_asynccnt), 5 loads issued back-to-back per thread.
//  - replicate-pad via clamped per-lane addresses (free on per-lane VGPR
//    addressing), separable 5-tap from LDS, coalesced stores.
//  - kernels templated on (H,W) so all tile/index math is shifts & masks.
// ---------------------------------------------------------------------------

#if defined(__gfx1250__)
  #if __has_builtin(__builtin_amdgcn_global_load_async_to_lds_b32)
    #define USE_ASYNC_LDS 1
  #else
    #define USE_ASYNC_LDS 0
    #warning "gfx1250: async global->LDS builtin unavailable; using fallback"
  #endif
  #if __has_builtin(__builtin_amdgcn_s_wait_asynccnt)
    #define HAVE_WAIT_ASYNC 1
  #else
    #define HAVE_WAIT_ASYNC 0
  #endif
#else
  #define USE_ASYNC_LDS 0
  #define HAVE_WAIT_ASYNC 0
#endif

#if USE_ASYNC_LDS
typedef __attribute__((address_space(1))) int* gas_int_ptr;  // global (AS1)
typedef __attribute__((address_space(3))) int* las_int_ptr;  // LDS    (AS3)
#endif

namespace {
constexpr int NT  = 256;          // 8 wave32 per workgroup
constexpr int TW  = 32;           // output tile width
constexpr int TH  = 8;            // output tile height
constexpr int IW  = 2 * TW + 3;   // 67 input cols (stride-2 + 5-tap halo)
constexpr int IH  = 2 * TH + 3;   // 19 input rows
constexpr int LST = IW + 1;       // 68: LDS row stride (bank padding)
constexpr int NC  = 16 * 3;       // batch * channels
constexpr int NELEM = IH * IW;    // 1273 staged elements per tile
}

template <int H, int W, bool FUSE_COPY>
__global__ __launch_bounds__(NT) void pyr_reduce(const float* __restrict__ src,
                                                 float* __restrict__ copy_dst,
                                                 float* __restrict__ dst)
{
    __shared__ float tile[IH * LST];

    constexpr int H2 = H / 2;
    constexpr int W2 = W / 2;
    constexpr unsigned TX = W2 / TW;   // 16 / 8 / 4  (pow2 -> masks)
    constexpr unsigned TY = H2 / TH;   // 64 / 32 / 16 (pow2 -> shifts)

    unsigned b = blockIdx.x;
    const int tx = (int)(b % TX); b /= TX;
    const int ty = (int)(b % TY); b /= TY;
    const int nc = (int)b;

    const float* __restrict__ s = src + (size_t)nc * (size_t)(H * W);
    float*       __restrict__ d = dst + (size_t)nc * (size_t)(H2 * W2);

    const int ox0 = tx * TW;
    const int oy0 = ty * TH;
    const int ix0 = 2 * ox0 - 2;    // leftmost input col (pre-clamp)
    const int iy0 = 2 * oy0 - 2;    // topmost  input row (pre-clamp)
    const int t   = (int)threadIdx.x;

    // Stage the 67x19 haloed tile into LDS: 5 async loads per thread,
    // issued back-to-back (tail indices clamped -> benign duplicate load).
#pragma unroll
    for (int k = 0; k < (NELEM + NT - 1) / NT; ++k) {
        int i = t + k * NT;
        if (i >= NELEM) i = NELEM - 1;
        const int r = i / IW;
        const int c = i - r * IW;
        int gy = iy0 + r; gy = gy < 0 ? 0 : (gy > H - 1 ? H - 1 : gy);
        int gx = ix0 + c; gx = gx < 0 ? 0 : (gx > W - 1 ? W - 1 : gx);
        const float* gp = s + (size_t)gy * (size_t)W + (size_t)gx;
        float*       lp = &tile[r * LST + c];
#if USE_ASYNC_LDS
        __builtin_amdgcn_global_load_async_to_lds_b32(
            (gas_int_ptr)gp, (las_int_ptr)lp, 0, 0);
#else
        *lp = *gp;
#endif
    }

#if USE_ASYNC_LDS
  #if HAVE_WAIT_ASYNC
    __builtin_amdgcn_s_wait_asynccnt(0);
  #else
    asm volatile("s_wait_asynccnt 0" ::: "memory");
  #endif
#endif
    __syncthreads();

    // Fused level-0 copy: the tile interior (local [2..17]x[2..65]) holds the
    // raw, unclamped input region 2*oy0..2*oy0+15 x 2*ox0..2*ox0+63.
    if constexpr (FUSE_COPY) {
        float* __restrict__ c0 = copy_dst + (size_t)nc * (size_t)(H * W)
                               + (size_t)(2 * oy0) * (size_t)W + (size_t)(2 * ox0);
#pragma unroll
        for (int k = 0; k < (2 * TW * 2 * TH) / NT; ++k) {
            const int i   = t + k * NT;        // 0..1023
            const int row = i >> 6;            // / 64
            const int col = i & 63;
            c0[(size_t)row * (size_t)W + (size_t)col] =
                tile[(row + 2) * LST + (col + 2)];
        }
    }

    // One output per thread: 32x8 tile. Separable 5-tap (1,4,6,4,1)/16.
    const int lx = t % TW;
    const int ly = t / TW;

    const float kv0 = 0.0625f, kv1 = 0.25f, kv2 = 0.375f;
    const float kv[5] = {kv0, kv1, kv2, kv1, kv0};

    float acc = 0.0f;
#pragma unroll
    for (int ky = 0; ky < 5; ++ky) {
        const float* rp = &tile[(2 * ly + ky) * LST + 2 * lx];
        float rowacc = kv[0] * rp[0];
        rowacc = fmaf(kv[1], rp[1], rowacc);
        rowacc = fmaf(kv[2], rp[2], rowacc);
        rowacc = fmaf(kv[3], rp[3], rowacc);
        rowacc = fmaf(kv[4], rp[4], rowacc);
        acc = fmaf(kv[ky], rowacc, acc);
    }

    d[(size_t)(oy0 + ly) * (size_t)W2 + (size_t)(ox0 + lx)] = acc;
}

extern "C" void kernel_launch(void* const* d_in, const int* in_sizes, int n_in,
                              void* d_out, int out_size, void* d_ws, size_t ws_size,
                              hipStream_t stream)
{
    (void)in_sizes; (void)n_in; (void)out_size; (void)d_ws; (void)ws_size;

    const float* im  = (const float*)d_in[0];
    float*       out = (float*)d_out;

    // Flat output layout (reference returns (L0, L1, L2, L3)):
    const size_t n0   = (size_t)16 * 3 * 1024 * 1024;  // 50,331,648
    const size_t off1 = n0;                            // L1: 16*3*512*512
    const size_t off2 = off1 + n0 / 4;                 // L2: 16*3*256*256
    const size_t off3 = off2 + n0 / 16;                // L3: 16*3*128*128

    // L0 copy is fused into the first reduction (single pass over the input).
    {
        constexpr unsigned blocks = (512 / TW) * (1024 / 2 / TH) * NC; // 49152
        pyr_reduce<1024, 1024, true><<<blocks, NT, 0, stream>>>(im, out, out + off1);
    }
    {
        constexpr unsigned blocks = (256 / TW) * (512 / 2 / TH) * NC;  // 12288
        pyr_reduce<512, 512, false><<<blocks, NT, 0, stream>>>(out + off1, nullptr, out + off2);
    }
    {
        constexpr unsigned blocks = (128 / TW) * (256 / 2 / TH) * NC;  // 3072
        pyr_reduce<256, 256, false><<<blocks, NT, 0, stream>>>(out + off2, nullptr, out + off3);
    }
}